// Multi_Behavior_Graph_Base_47579647705297
// MI455X (gfx1250) — compile-verified
//
#include <hip/hip_runtime.h>
#include <hip/hip_bf16.h>
#include <math.h>

#define EPSV 1e-8f
#define REGV 1e-4f

#define NU    100000
#define NI    30000
#define ND    64
#define NB    3
#define NEUI  500000
#define NEIG  400000
#define NETR  1000000
#define NBATCH 2048
#define NK    20

typedef __attribute__((ext_vector_type(2))) float v2f;
typedef __attribute__((ext_vector_type(8))) float v8f;

// ---------------------------------------------------------------- fills
__global__ void fill_f32(float* p, size_t n, float v) {
  size_t i = (size_t)blockIdx.x * blockDim.x + threadIdx.x;
  size_t stride = (size_t)gridDim.x * blockDim.x;
  for (; i < n; i += stride) p[i] = v;
}

__global__ void fill_i32(int* p, size_t n, int v) {
  size_t i = (size_t)blockIdx.x * blockDim.x + threadIdx.x;
  size_t stride = (size_t)gridDim.x * blockDim.x;
  for (; i < n; i += stride) p[i] = v;
}

// ------------------------------------------------- user -> slot map (dedupe)
__global__ void build_map(const int* __restrict__ user, int* __restrict__ mapU) {
  int bi = blockIdx.x * blockDim.x + threadIdx.x;
  if (bi < NBATCH) atomicMin(&mapU[user[bi]], bi);
}

// ------------------------------------- item-item graph atomic segment sum
// tmpII[b][r][:] += item_emb[c][:] for each ig edge (r,c) of behavior b
__global__ void ig_scatter(const int* __restrict__ ig_rows, const int* __restrict__ ig_cols,
                           const float* __restrict__ item_emb, float* __restrict__ tmpII) {
  long t = (long)blockIdx.x * blockDim.x + threadIdx.x;
  const long total = (long)NB * NEIG * 16;
  if (t >= total) return;
  int  part = (int)(t & 15);        // 16 parts x float4 = 64 floats
  long e    = t >> 4;
  int  b    = (int)(e / NEIG);
  long eb   = e % NEIG;
  int r = ig_rows[(long)b * NEIG + eb];
  int c = ig_cols[(long)b * NEIG + eb];
  float4 v = *(const float4*)(item_emb + (long)c * ND + part * 4);
  float* dst = tmpII + ((long)b * NI + r) * ND + part * 4;
  atomicAdd(dst + 0, v.x); atomicAdd(dst + 1, v.y);
  atomicAdd(dst + 2, v.z); atomicAdd(dst + 3, v.w);
}

// divide by (ig_deg + eps)
__global__ void scale_ii(float* __restrict__ tmpII, const float* __restrict__ ig_deg) {
  long t = (long)blockIdx.x * blockDim.x + threadIdx.x;
  const long total = (long)NB * NI * ND;
  if (t >= total) return;
  long row = t >> 6;                 // b*NI + i
  tmpII[t] = tmpII[t] / (ig_deg[row] + EPSV);
}

// ------------------------------------------------------- WMMA f32 GEMM
// C[M,N] = A[M,K] @ B[K,N]  (row major, ld* strides). Optional epilogue:
// C[m, n<64] += addRow[(addIdx? addIdx[m] : m)*64 + n]
// One wave per 16x16 tile, V_WMMA_F32_16X16X4_F32 over K/4 steps.
__global__ void gemm_wmma_f32(const float* __restrict__ A, int lda,
                              const float* __restrict__ B, int ldb,
                              float* __restrict__ C, int ldc,
                              int M, int N, int Kd,
                              const float* __restrict__ addRow,
                              const int* __restrict__ addIdx) {
  int wave = (blockIdx.x * blockDim.x + threadIdx.x) >> 5;   // uniform per wave
  int lane = threadIdx.x & 31;
  int tilesN = N >> 4;
  int totalTiles = (M >> 4) * tilesN;
  if (wave >= totalTiles) return;                            // whole-wave exit (EXEC stays all-1)
  int tm = (wave / tilesN) << 4;
  int tn = (wave % tilesN) << 4;
  int half = lane >> 4;            // 0 or 1
  int mn   = lane & 15;

  v8f acc = {};
  const float* arow = A + (long)(tm + mn) * lda;
  for (int k0 = 0; k0 < Kd; k0 += 4) {
    int kb = k0 + 2 * half;
    v2f a, bb;
    a.x  = arow[kb];
    a.y  = arow[kb + 1];
    bb.x = B[(long)kb * ldb + tn + mn];
    bb.y = B[(long)(kb + 1) * ldb + tn + mn];
    acc = __builtin_amdgcn_wmma_f32_16x16x4_f32(
        /*neg_a=*/false, a, /*neg_b=*/false, bb,
        /*c_mod=*/(short)0, acc, /*reuse_a=*/false, /*reuse_b=*/false);
  }
  for (int j = 0; j < 8; ++j) {
    int row = tm + j + 8 * half;
    int col = tn + mn;
    float v = acc[j];
    if (addRow != nullptr && col < ND) {
      int er = addIdx ? addIdx[row] : row;
      v += addRow[(long)er * ND + col];
    }
    C[(long)row * ldc + col] = v;
  }
}

// ------------------- filtered UI-edge scatter into per-slot n2u rows
__global__ void ui_scatter(const int* __restrict__ ui_rows, const int* __restrict__ ui_cols,
                           const int* __restrict__ mapU,
                           const float* __restrict__ item_emb, const float* __restrict__ ii,
                           float* __restrict__ n2u) {
  long t = (long)blockIdx.x * blockDim.x + threadIdx.x;
  const long total = (long)NB * NEUI;
  if (t >= total) return;
  int  b = (int)(t / NEUI);
  long e = t % NEUI;
  int u = ui_rows[(long)b * NEUI + e];
  int s = mapU[u];
  if (s >= NBATCH) return;                      // user not in batch -> skip edge
  int c = ui_cols[(long)b * NEUI + e];
  const float* src0 = item_emb + (long)c * ND;
  const float* src1 = ii + ((long)b * NI + c) * ND;
  float* dst = n2u + ((long)b * NBATCH + s) * (2 * ND);
  for (int j = 0; j < ND; j += 4) {
    float4 v = *(const float4*)(src0 + j);
    atomicAdd(dst + j + 0, v.x); atomicAdd(dst + j + 1, v.y);
    atomicAdd(dst + j + 2, v.z); atomicAdd(dst + j + 3, v.w);
  }
  for (int j = 0; j < ND; j += 4) {
    float4 v = *(const float4*)(src1 + j);
    atomicAdd(dst + ND + j + 0, v.x); atomicAdd(dst + ND + j + 1, v.y);
    atomicAdd(dst + ND + j + 2, v.z); atomicAdd(dst + ND + j + 3, v.w);
  }
}

// per-slot degree normalize + behavior-mix into user_feature
__global__ void slot_norm(const int* __restrict__ user, const int* __restrict__ mapU,
                          const float* __restrict__ ubd, const float* __restrict__ bw,
                          float* __restrict__ n2u, float* __restrict__ ufeat) {
  int t = blockIdx.x * blockDim.x + threadIdx.x;       // NBATCH * 128
  if (t >= NBATCH * 128) return;
  int s = t >> 7;
  int j = t & 127;
  int u = user[s];
  if (mapU[u] != s) return;                            // only canonical slots
  float d0 = ubd[u * NB + 0], d1 = ubd[u * NB + 1], d2 = ubd[u * NB + 2];
  float w0 = bw[0], w1 = bw[1], w2 = bw[2];
  float tot = d0 * w0 + d1 * w1 + d2 * w2 + EPSV;
  float deg[NB]  = {d0, d1, d2};
  float mixw[NB] = {d0 * w0 / tot, d1 * w1 / tot, d2 * w2 / tot};
  float acc = 0.f;
  for (int b = 0; b < NB; ++b) {
    long idx = ((long)b * NBATCH + s) * (2 * ND) + j;
    float v = n2u[idx] / (deg[b] + EPSV);
    n2u[idx] = v;
    if (j < ND) acc += mixw[b] * v;                    // un == n2u[:, :D]
  }
  if (j < ND) ufeat[(long)s * ND + j] = acc;
}

// item_feature = train^T @ user_emb   (atomic segment sum over 1M edges)
__global__ void tr_scatter(const int* __restrict__ tr_rows, const int* __restrict__ tr_cols,
                           const float* __restrict__ user_emb, float* __restrict__ ifeat) {
  long t = (long)blockIdx.x * blockDim.x + threadIdx.x;
  const long total = (long)NETR * 16;
  if (t >= total) return;
  int  part = (int)(t & 15);
  long e    = t >> 4;
  int r = tr_rows[e];           // user
  int c = tr_cols[e];           // item
  float4 v = *(const float4*)(user_emb + (long)r * ND + part * 4);
  float* dst = ifeat + (long)c * ND + part * 4;
  atomicAdd(dst + 0, v.x); atomicAdd(dst + 1, v.y);
  atomicAdd(dst + 2, v.z); atomicAdd(dst + 3, v.w);
}

// Wcomb = side_W + [0 | W]  (folds concat(emb, X@W)+X@side_W into one GEMM)
__global__ void build_wcomb(const float* __restrict__ W, const float* __restrict__ userW,
                            const float* __restrict__ itemW,
                            float* __restrict__ wcu, float* __restrict__ wci) {
  int t = blockIdx.x * blockDim.x + threadIdx.x;   // 64*128
  if (t >= ND * 2 * ND) return;
  int r = t >> 7;
  int c = t & 127;
  float add = (c >= ND) ? W[r * ND + (c - ND)] : 0.f;
  wcu[t] = userW[t] + add;
  wci[t] = itemW[t] + add;
}

// fused scoring: score1, score2, gate, L2 (block-reduced)
__global__ void score_kernel(const int* __restrict__ user, const int* __restrict__ item,
                             const int* __restrict__ mapU,
                             const float* __restrict__ uf, const float* __restrict__ itf,
                             const float* __restrict__ i2u, const float* __restrict__ item_emb,
                             const float* __restrict__ ii, const float* __restrict__ sigW,
                             float* __restrict__ out) {
  __shared__ float red[256];
  int t = blockIdx.x * blockDim.x + threadIdx.x;
  float l2 = 0.f;
  if (t < NBATCH * NK) {
    int bi = t / NK;
    int s  = mapU[user[bi]];
    int it = item[t];
    const float* ufr = uf  + (long)s  * (2 * ND);
    const float* itr = itf + (long)it * (2 * ND);
    float s1 = 0.f, sw = 0.f, nu = 0.f, ni = 0.f;
    for (int j = 0; j < 2 * ND; ++j) {
      float a = ufr[j], b = itr[j];
      s1 += a * b;
      sw += a * sigW[j];
      nu += a * a;
      ni += b * b;
    }
    float s2 = 0.f;
    const float* iemb = item_emb + (long)it * ND;
    for (int b = 0; b < NB; ++b) {
      const float* i2  = i2u + ((long)b * NBATCH + s) * (2 * ND);
      const float* iib = ii  + ((long)b * NI + it) * ND;
      for (int j = 0; j < ND; ++j)
        s2 += i2[j] * iemb[j] + i2[ND + j] * iib[j];
    }
    s2 *= (1.0f / (float)NB);
    float gate = 1.f / (1.f + expf(-sw));
    out[t] = s1 * gate + s2 * (1.f - gate);
    l2 = REGV * (nu + ni);
  }
  red[threadIdx.x] = l2;
  __syncthreads();
  for (int off = 128; off > 0; off >>= 1) {
    if (threadIdx.x < off) red[threadIdx.x] += red[threadIdx.x + off];
    __syncthreads();
  }
  if (threadIdx.x == 0) atomicAdd(&out[NBATCH * NK], red[0]);
}

// ----------------------------------------------------------------- launch
extern "C" void kernel_launch(void* const* d_in, const int* in_sizes, int n_in,
                              void* d_out, int out_size, void* d_ws, size_t ws_size,
                              hipStream_t stream) {
  const float* user_emb = (const float*)d_in[0];
  const float* item_emb = (const float*)d_in[1];
  const float* ubd      = (const float*)d_in[2];
  const float* bw       = (const float*)d_in[3];
  const float* ig_deg   = (const float*)d_in[4];
  const float* ibW      = (const float*)d_in[5];   // [B,128,128]
  const float* ipW      = (const float*)d_in[6];   // [B,64,64]
  const float* W        = (const float*)d_in[7];   // [64,64]
  const float* userW    = (const float*)d_in[8];   // [64,128]
  const float* itemW    = (const float*)d_in[9];   // [64,128]
  const float* sigW     = (const float*)d_in[10];  // [128,1]
  const int* ui_rows    = (const int*)d_in[11];
  const int* ui_cols    = (const int*)d_in[12];
  const int* ig_rows    = (const int*)d_in[13];
  const int* ig_cols    = (const int*)d_in[14];
  const int* tr_rows    = (const int*)d_in[15];
  const int* tr_cols    = (const int*)d_in[16];
  const int* user       = (const int*)d_in[17];    // [2048]
  const int* item       = (const int*)d_in[18];    // [2048,20]
  float* out = (float*)d_out;

  float* ws = (float*)d_ws;
  size_t off = 0;
  float* tmpII = ws + off;  off += (size_t)NB * NI * ND;        // zeroed
  float* n2u   = ws + off;  off += (size_t)NB * NBATCH * 2 * ND;// zeroed
  float* ufeat = ws + off;  off += (size_t)NBATCH * ND;         // zeroed
  float* ifeat = ws + off;  off += (size_t)NI * ND;             // zeroed
  size_t zeroCount = off;
  float* ii    = ws + off;  off += (size_t)NB * NI * ND;
  float* i2u   = ws + off;  off += (size_t)NB * NBATCH * 2 * ND;
  float* uf    = ws + off;  off += (size_t)NBATCH * 2 * ND;
  float* itf   = ws + off;  off += (size_t)NI * 2 * ND;
  float* wcu   = ws + off;  off += (size_t)ND * 2 * ND;
  float* wci   = ws + off;  off += (size_t)ND * 2 * ND;
  int*   mapU  = (int*)(ws + off);

  // init
  fill_f32<<<2048, 256, 0, stream>>>(ws, zeroCount, 0.f);
  fill_f32<<<1, 32, 0, stream>>>(out + NBATCH * NK, 1, 0.f);
  fill_i32<<<512, 256, 0, stream>>>(mapU, (size_t)NU, 0x7FFFFFFF);

  build_map<<<(NBATCH + 255) / 256, 256, 0, stream>>>(user, mapU);

  { long total = (long)NB * NEIG * 16;
    ig_scatter<<<(int)((total + 255) / 256), 256, 0, stream>>>(ig_rows, ig_cols, item_emb, tmpII); }

  { long total = (long)NB * NI * ND;
    scale_ii<<<(int)((total + 255) / 256), 256, 0, stream>>>(tmpII, ig_deg); }

  // ii[b] = (segsum/deg) @ item_propagate_W[b]   [30000x64]@[64x64]
  for (int b = 0; b < NB; ++b) {
    int tiles = (NI / 16) * (ND / 16);
    gemm_wmma_f32<<<(tiles * 32 + 255) / 256, 256, 0, stream>>>(
        tmpII + (size_t)b * NI * ND, ND, ipW + (size_t)b * ND * ND, ND,
        ii + (size_t)b * NI * ND, ND, NI, ND, ND, nullptr, nullptr);
  }

  { long total = (long)NB * NEUI;
    ui_scatter<<<(int)((total + 255) / 256), 256, 0, stream>>>(
        ui_rows, ui_cols, mapU, item_emb, ii, n2u); }

  slot_norm<<<(NBATCH * 128 + 255) / 256, 256, 0, stream>>>(user, mapU, ubd, bw, n2u, ufeat);

  // i2u[b] = n2u[b] @ item_behaviour_W[b]   [2048x128]@[128x128]
  for (int b = 0; b < NB; ++b) {
    int tiles = (NBATCH / 16) * (2 * ND / 16);
    gemm_wmma_f32<<<(tiles * 32 + 255) / 256, 256, 0, stream>>>(
        n2u + (size_t)b * NBATCH * 2 * ND, 2 * ND, ibW + (size_t)b * 2 * ND * 2 * ND, 2 * ND,
        i2u + (size_t)b * NBATCH * 2 * ND, 2 * ND, NBATCH, 2 * ND, 2 * ND, nullptr, nullptr);
  }

  { long total = (long)NETR * 16;
    tr_scatter<<<(int)((total + 255) / 256), 256, 0, stream>>>(tr_rows, tr_cols, user_emb, ifeat); }

  build_wcomb<<<(ND * 2 * ND + 255) / 256, 256, 0, stream>>>(W, userW, itemW, wcu, wci);

  // itf = item_feature @ (item_W + [0|W]) + [item_emb | 0]   [30000x64]@[64x128]
  { int tiles = (NI / 16) * (2 * ND / 16);
    gemm_wmma_f32<<<(tiles * 32 + 255) / 256, 256, 0, stream>>>(
        ifeat, ND, wci, 2 * ND, itf, 2 * ND, NI, 2 * ND, ND, item_emb, nullptr); }

  // uf = user_feature @ (user_W + [0|W]) + [user_emb[user] | 0]   [2048x64]@[64x128]
  { int tiles = (NBATCH / 16) * (2 * ND / 16);
    gemm_wmma_f32<<<(tiles * 32 + 255) / 256, 256, 0, stream>>>(
        ufeat, ND, wcu, 2 * ND, uf, 2 * ND, NBATCH, 2 * ND, ND, user_emb, user); }

  score_kernel<<<(NBATCH * NK) / 256, 256, 0, stream>>>(
      user, item, mapU, uf, itf, i2u, item_emb, ii, sigW, out);
}